// CriteriaDVHLoss_6640019440296
// MI455X (gfx1250) — compile-verified
//
#include <hip/hip_runtime.h>
#include <hip/hip_bf16.h>

// ---------------- problem constants ----------------
#define BATCH 8
#define NROW (128 * 128 * 128)      // 2,097,152 elements per batch row
#define DOSE_MAXF 52.0f
#define NBINS 2048                   // value bins over [0, 52]
#define THREADS 256
#define ELEMS_PER_THREAD 64          // 16 x float4 per thread
#define BLOCKS_PER_ROW 128           // 128*256*64 == NROW
#define NSEL 6                       // lo/hi ranks for 3 quantiles
#define CAP 4096                     // candidate buffer per (arr,row,sel); expect ~512

typedef __attribute__((ext_vector_type(2))) float v2f;
typedef __attribute__((ext_vector_type(8))) float v8f;

__device__ __forceinline__ int vbin(float p) {
  int b = (int)(p * (NBINS / DOSE_MAXF));
  b = b < 0 ? 0 : b;
  return b > (NBINS - 1) ? (NBINS - 1) : b;
}

// ------------------------------------------------------------------
// Pass A: one streaming read of everything.
//  - per-row 2048-bin value histograms of masked pred / target (LDS -> global)
//  - per-block partial sums: [n_ptv, cnt0..2, sumP0..2, sumG0..2, pad] (16 slots)
//  - per-row OAR maxes via atomicMax on uint (values are >= 0)
// ------------------------------------------------------------------
__global__ void k_hist(const float* __restrict__ pred,
                       const float* __restrict__ target,
                       const unsigned char* __restrict__ pm,
                       const unsigned char* __restrict__ o0,
                       const unsigned char* __restrict__ o1,
                       const unsigned char* __restrict__ o2,
                       unsigned int* __restrict__ hist0,     // [2][BATCH][NBINS]
                       float* __restrict__ partials,         // [BATCH][BLOCKS_PER_ROW][16]
                       unsigned int* __restrict__ maxes) {   // [BATCH][6]  (pmax0..2, gmax0..2)
  __shared__ unsigned int lhist[2 * NBINS];
  __shared__ float sred[THREADS];

  const int row = blockIdx.y;
  const int tid = threadIdx.x;
  for (int i = tid; i < 2 * NBINS; i += THREADS) lhist[i] = 0;
  __syncthreads();

  const size_t base = (size_t)row * NROW + (size_t)blockIdx.x * (THREADS * ELEMS_PER_THREAD);
  const float4* p4 = (const float4*)(pred + base);
  const float4* g4 = (const float4*)(target + base);
  const uchar4* m4 = (const uchar4*)(pm + base);
  const uchar4* a4 = (const uchar4*)(o0 + base);
  const uchar4* b4 = (const uchar4*)(o1 + base);
  const uchar4* c4 = (const uchar4*)(o2 + base);

  float nptv = 0.0f;
  float cnt[3] = {0.f, 0.f, 0.f};
  float sp[3] = {0.f, 0.f, 0.f};
  float sg[3] = {0.f, 0.f, 0.f};
  float mp[3] = {0.f, 0.f, 0.f};   // values >= 0; gated by cnt>0 downstream
  float mg[3] = {0.f, 0.f, 0.f};

  for (int it = 0; it < ELEMS_PER_THREAD / 4; ++it) {
    const int idx = tid + it * THREADS;
    __builtin_prefetch(&p4[idx + THREADS], 0, 0);   // global_prefetch_b8
    float pv[4], gv[4];
    *(float4*)pv = p4[idx];
    *(float4*)gv = g4[idx];
    uchar4 mB = m4[idx], aB = a4[idx], bB = b4[idx], cB = c4[idx];
    unsigned char mA[4] = {mB.x, mB.y, mB.z, mB.w};
    unsigned char aA[4] = {aB.x, aB.y, aB.z, aB.w};
    unsigned char bA[4] = {bB.x, bB.y, bB.z, bB.w};
    unsigned char cA[4] = {cB.x, cB.y, cB.z, cB.w};
#pragma unroll
    for (int e = 0; e < 4; ++e) {
      const float p = pv[e] * DOSE_MAXF;
      const float g = gv[e] * DOSE_MAXF;
      if (mA[e]) {
        nptv += 1.0f;
        atomicAdd(&lhist[vbin(p)], 1u);
        atomicAdd(&lhist[NBINS + vbin(g)], 1u);
      }
      if (aA[e]) { cnt[0] += 1.f; sp[0] += p; sg[0] += g; mp[0] = fmaxf(mp[0], p); mg[0] = fmaxf(mg[0], g); }
      if (bA[e]) { cnt[1] += 1.f; sp[1] += p; sg[1] += g; mp[1] = fmaxf(mp[1], p); mg[1] = fmaxf(mg[1], g); }
      if (cA[e]) { cnt[2] += 1.f; sp[2] += p; sg[2] += g; mp[2] = fmaxf(mp[2], p); mg[2] = fmaxf(mg[2], g); }
    }
  }

  // ---- block reduction of the 10 sums -> partials[row][block][0..9]
  float vals[10] = {nptv, cnt[0], cnt[1], cnt[2], sp[0], sp[1], sp[2], sg[0], sg[1], sg[2]};
  float* pout = partials + ((size_t)row * BLOCKS_PER_ROW + blockIdx.x) * 16;
  for (int q = 0; q < 10; ++q) {
    sred[tid] = vals[q];
    __syncthreads();
    for (int s = THREADS / 2; s > 0; s >>= 1) {
      if (tid < s) sred[tid] += sred[tid + s];
      __syncthreads();
    }
    if (tid == 0) pout[q] = sred[0];
    __syncthreads();
  }
  if (tid == 0) {
    for (int q = 10; q < 16; ++q) pout[q] = 0.0f;
  }

  // ---- block max reduction -> atomicMax on uint (monotone for non-negative floats)
  float mvals[6] = {mp[0], mp[1], mp[2], mg[0], mg[1], mg[2]};
  for (int q = 0; q < 6; ++q) {
    sred[tid] = mvals[q];
    __syncthreads();
    for (int s = THREADS / 2; s > 0; s >>= 1) {
      if (tid < s) sred[tid] = fmaxf(sred[tid], sred[tid + s]);
      __syncthreads();
    }
    if (tid == 0) atomicMax(&maxes[row * 6 + q], __float_as_uint(sred[0]));
    __syncthreads();
  }

  // ---- flush LDS histograms
  for (int i = tid; i < 2 * NBINS; i += THREADS) {
    unsigned int v = lhist[i];
    if (v) {
      const int arr = (i < NBINS) ? 0 : 1;
      atomicAdd(&hist0[((size_t)arr * BATCH + row) * NBINS + (i & (NBINS - 1))], v);
    }
  }
}

// ------------------------------------------------------------------
// WMMA reduction of partial sums: totals[row][j] = sum_b partials[row][b][j]
// A = all-ones 16x4 (layout-invariant), B tiles = 4 partial rows x 16 quantities.
// f32 WMMA is exact for integer counts < 2^24.
// One wave32 per row.  Emits v_wmma_f32_16x16x4_f32.
// ------------------------------------------------------------------
__global__ void k_reduce_wmma(const float* __restrict__ partials,
                              float* __restrict__ totals) {   // [BATCH][16]
  const int row = blockIdx.x;
  const int lane = threadIdx.x;          // 0..31, full wave -> EXEC all ones
  const int j = lane & 15;
  const int half = lane >> 4;

  v8f acc = {};
  v2f a;
  a.x = 1.0f;
  a.y = 1.0f;

  const float* base = partials + (size_t)row * BLOCKS_PER_ROW * 16;
  for (int c = 0; c < BLOCKS_PER_ROW / 4; ++c) {
    const int r0 = c * 4 + half * 2;
    v2f b;
    b.x = base[(r0 + 0) * 16 + j];
    b.y = base[(r0 + 1) * 16 + j];
    acc = __builtin_amdgcn_wmma_f32_16x16x4_f32(false, a, false, b, (short)0, acc,
                                                false, false);
  }
  if (lane < 16) totals[row * 16 + j] = acc[0];  // C/D VGPR0, lanes 0-15 = (M=0, N=lane)
}

// ------------------------------------------------------------------
// Scan: for each (arr,row) find the bin + within-bin rank of the 6 order stats.
// ------------------------------------------------------------------
__global__ void k_scan(const unsigned int* __restrict__ hist0,
                       const float* __restrict__ totals,
                       int* __restrict__ selBin,    // [2][BATCH][6]
                       int* __restrict__ selRank) { // [2][BATCH][6]
  const int t = threadIdx.x;
  if (t >= 2 * BATCH) return;
  const int arr = t >> 3;
  const int row = t & 7;
  const float n = totals[row * 16];
  const float qs[3] = {0.99f, 0.95f, 0.01f};

  long ranks[NSEL];
  for (int qi = 0; qi < 3; ++qi) {
    const float pos = fmaxf(qs[qi] * (n - 1.0f), 0.0f);
    ranks[2 * qi] = (long)floorf(pos);
    ranks[2 * qi + 1] = (long)ceilf(pos);
  }

  const unsigned int* h = hist0 + ((size_t)arr * BATCH + row) * NBINS;
  for (int s = 0; s < NSEL; ++s) {
    int bin = -1, rib = 0;
    if (n > 0.0f) {
      long cum = 0;
      for (int b = 0; b < NBINS; ++b) {
        const long c = (long)h[b];
        if (ranks[s] < cum + c) { bin = b; rib = (int)(ranks[s] - cum); break; }
        cum += c;
      }
    }
    selBin[((size_t)arr * BATCH + row) * NSEL + s] = bin;
    selRank[((size_t)arr * BATCH + row) * NSEL + s] = rib;
  }
}

// ------------------------------------------------------------------
// Scatter: second streaming pass (pred/target/ptv only; mostly L2-resident).
// Collect masked elements that land in a selected bin into small buffers.
// ------------------------------------------------------------------
__global__ void k_scatter(const float* __restrict__ pred,
                          const float* __restrict__ target,
                          const unsigned char* __restrict__ pm,
                          const int* __restrict__ selBin,
                          unsigned int* __restrict__ candCnt,  // [2*BATCH*NSEL]
                          float* __restrict__ candBuf) {       // [2*BATCH*NSEL][CAP]
  const int row = blockIdx.y;
  const int tid = threadIdx.x;

  int sb0[NSEL], sb1[NSEL];
  for (int s = 0; s < NSEL; ++s) {
    sb0[s] = selBin[((size_t)0 * BATCH + row) * NSEL + s];
    sb1[s] = selBin[((size_t)1 * BATCH + row) * NSEL + s];
  }

  const size_t base = (size_t)row * NROW + (size_t)blockIdx.x * (THREADS * ELEMS_PER_THREAD);
  const float4* p4 = (const float4*)(pred + base);
  const float4* g4 = (const float4*)(target + base);
  const uchar4* m4 = (const uchar4*)(pm + base);

  for (int it = 0; it < ELEMS_PER_THREAD / 4; ++it) {
    const int idx = tid + it * THREADS;
    float pv[4], gv[4];
    *(float4*)pv = p4[idx];
    *(float4*)gv = g4[idx];
    uchar4 mB = m4[idx];
    unsigned char mA[4] = {mB.x, mB.y, mB.z, mB.w};
#pragma unroll
    for (int e = 0; e < 4; ++e) {
      if (!mA[e]) continue;
      const float p = pv[e] * DOSE_MAXF;
      const float g = gv[e] * DOSE_MAXF;
      const int bp = vbin(p);
      const int bg = vbin(g);
      for (int s = 0; s < NSEL; ++s) {
        if (bp == sb0[s]) {
          const size_t sid = ((size_t)0 * BATCH + row) * NSEL + s;
          const unsigned int w = atomicAdd(&candCnt[sid], 1u);
          if (w < CAP) candBuf[sid * CAP + w] = p;
        }
        if (bg == sb1[s]) {
          const size_t sid = ((size_t)1 * BATCH + row) * NSEL + s;
          const unsigned int w = atomicAdd(&candCnt[sid], 1u);
          if (w < CAP) candBuf[sid * CAP + w] = g;
        }
      }
    }
  }
}

// ------------------------------------------------------------------
// Select: bitonic-sort each candidate buffer in LDS, pick the within-bin rank.
// ------------------------------------------------------------------
__global__ void k_select(const float* __restrict__ candBuf,
                         const unsigned int* __restrict__ candCnt,
                         const int* __restrict__ selBin,
                         const int* __restrict__ selRank,
                         float* __restrict__ quantVal) {  // [2*BATCH*NSEL]
  __shared__ float sd[CAP];
  const int sid = blockIdx.x;
  const int tid = threadIdx.x;
  const unsigned int cRaw = candCnt[sid];
  const int c = (int)(cRaw < CAP ? cRaw : CAP);

  for (int i = tid; i < CAP; i += THREADS)
    sd[i] = (i < c) ? candBuf[(size_t)sid * CAP + i] : __builtin_inff();
  __syncthreads();

  for (int k = 2; k <= CAP; k <<= 1) {
    for (int j = k >> 1; j > 0; j >>= 1) {
      for (int i = tid; i < CAP; i += THREADS) {
        const int ixj = i ^ j;
        if (ixj > i) {
          const bool asc = ((i & k) == 0);
          const float x = sd[i];
          const float y = sd[ixj];
          if (asc ? (x > y) : (x < y)) { sd[i] = y; sd[ixj] = x; }
        }
      }
      __syncthreads();
    }
  }

  if (tid == 0) {
    float v = 0.0f;
    if (selBin[sid] >= 0 && c > 0) {
      int r = selRank[sid];
      r = r < 0 ? 0 : (r >= c ? c - 1 : r);
      v = sd[r];
    }
    quantVal[sid] = v;
  }
}

// ------------------------------------------------------------------
// Finalize: assemble the scalar loss.
// ------------------------------------------------------------------
__global__ void k_final(const float* __restrict__ totals,
                        const unsigned int* __restrict__ maxes,
                        const float* __restrict__ quantVal,
                        float* __restrict__ out) {
  __shared__ float sl[BATCH];
  const int tid = threadIdx.x;
  if (tid < BATCH) {
    const int row = tid;
    const float n = totals[row * 16];
    float loss = 0.0f;
    const float qs[3] = {0.99f, 0.95f, 0.01f};
    for (int qi = 0; qi < 3; ++qi) {
      const float pos = fmaxf(qs[qi] * (n - 1.0f), 0.0f);
      const float frac = pos - floorf(pos);
      const float plo = quantVal[((size_t)0 * BATCH + row) * NSEL + 2 * qi];
      const float phi = quantVal[((size_t)0 * BATCH + row) * NSEL + 2 * qi + 1];
      const float glo = quantVal[((size_t)1 * BATCH + row) * NSEL + 2 * qi];
      const float ghi = quantVal[((size_t)1 * BATCH + row) * NSEL + 2 * qi + 1];
      const float pq = plo + (phi - plo) * frac;
      const float gq = glo + (ghi - glo) * frac;
      if (n > 0.0f) loss += fabsf(pq - gq);
    }
    for (int k = 0; k < 3; ++k) {
      const float cnt = totals[row * 16 + 1 + k];
      const float spk = totals[row * 16 + 4 + k];
      const float sgk = totals[row * 16 + 7 + k];
      const float pmax = __uint_as_float(maxes[row * 6 + k]);
      const float gmax = __uint_as_float(maxes[row * 6 + 3 + k]);
      const float denom = fmaxf(cnt, 1.0f);
      if (cnt > 0.0f) loss += fabsf(pmax - gmax) + fabsf(spk / denom - sgk / denom);
    }
    sl[row] = loss;
  }
  __syncthreads();
  if (tid == 0) {
    float s = 0.0f;
    for (int r = 0; r < BATCH; ++r) s += sl[r];
    out[0] = s / (float)BATCH;
  }
}

// ------------------------------------------------------------------
// Host launcher.
// Workspace layout (bytes):
//   hist0    @ 0        : 2*8*2048*4 = 131072
//   maxes    @ 131072   : 8*6*4      = 192
//   candCnt  @ 131264   : 96*4       = 384     (zeroed region ends at 131648)
//   totals   @ 131648   : 8*16*4     = 512
//   selBin   @ 132160   : 96*4       = 384
//   selRank  @ 132544   : 96*4       = 384
//   quantVal @ 132928   : 96*4       = 384
//   partials @ 133312   : 8*128*16*4 = 65536
//   candBuf  @ 198848   : 96*4096*4  = 1572864   (total ~1.77 MB)
// ------------------------------------------------------------------
extern "C" void kernel_launch(void* const* d_in, const int* in_sizes, int n_in,
                              void* d_out, int out_size, void* d_ws, size_t ws_size,
                              hipStream_t stream) {
  const float* pred = (const float*)d_in[0];
  const float* target = (const float*)d_in[1];
  const unsigned char* pm = (const unsigned char*)d_in[2];
  const unsigned char* o0 = (const unsigned char*)d_in[3];
  const unsigned char* o1 = (const unsigned char*)d_in[4];
  const unsigned char* o2 = (const unsigned char*)d_in[5];

  char* ws = (char*)d_ws;
  unsigned int* hist0 = (unsigned int*)(ws + 0);
  unsigned int* maxes = (unsigned int*)(ws + 131072);
  unsigned int* candCnt = (unsigned int*)(ws + 131264);
  float* totals = (float*)(ws + 131648);
  int* selBin = (int*)(ws + 132160);
  int* selRank = (int*)(ws + 132544);
  float* quantVal = (float*)(ws + 132928);
  float* partials = (float*)(ws + 133312);
  float* candBuf = (float*)(ws + 198848);

  // zero the atomic accumulators (hist0 + maxes + candCnt) every launch
  hipMemsetAsync(d_ws, 0, 131648, stream);

  dim3 gStream(BLOCKS_PER_ROW, BATCH);
  k_hist<<<gStream, THREADS, 0, stream>>>(pred, target, pm, o0, o1, o2,
                                          hist0, partials, maxes);
  k_reduce_wmma<<<BATCH, 32, 0, stream>>>(partials, totals);
  k_scan<<<1, 32, 0, stream>>>(hist0, totals, selBin, selRank);
  k_scatter<<<gStream, THREADS, 0, stream>>>(pred, target, pm, selBin, candCnt, candBuf);
  k_select<<<2 * BATCH * NSEL, THREADS, 0, stream>>>(candBuf, candCnt, selBin, selRank,
                                                     quantVal);
  k_final<<<1, 32, 0, stream>>>(totals, maxes, quantVal, (float*)d_out);
}